// GINELayer_19550691131956
// MI455X (gfx1250) — compile-verified
//
#include <hip/hip_runtime.h>
#include <stdint.h>

#define HIDDEN  64
#define NNODES  100000
#define NEDGES  1600000

typedef __attribute__((ext_vector_type(2))) float v2f;
typedef __attribute__((ext_vector_type(8))) float v8f;

// ---------------- kernel 0: zero the aggregation buffer ----------------
__global__ void gine_zero_kernel(float* __restrict__ p, int n4) {
    int i = blockIdx.x * blockDim.x + threadIdx.x;
    if (i < n4) {
        float4 z; z.x = 0.f; z.y = 0.f; z.z = 0.f; z.w = 0.f;
        ((float4*)p)[i] = z;
    }
}

// ---------------- kernel 1: edge message + scatter-add -----------------
// One wave32 per edge; lane owns channels (2*lane, 2*lane+1).
// Software-pipelined: next edge's indices/attrs load + x-row prefetch
// (global_prefetch_b8) overlap the current edge's compute + atomics.
__global__ void __launch_bounds__(256)
gine_edge_kernel(const float* __restrict__ x,
                 const long long* __restrict__ ei,   // [2, E] int64
                 const float* __restrict__ ea,       // [E, 4]
                 const float* __restrict__ We,       // [4, 64]
                 const float* __restrict__ be,       // [64]
                 float* __restrict__ agg)            // [N, 64]
{
    const int lane = threadIdx.x & 31;
    const int wv   = threadIdx.x >> 5;
    const int nwpb = blockDim.x >> 5;
    const int wave = blockIdx.x * nwpb + wv;
    const int nwv  = gridDim.x * nwpb;
    const int c0   = lane * 2;

    // hoist per-lane slice of We/be out of the edge loop
    float w00 = We[0 * HIDDEN + c0], w01 = We[0 * HIDDEN + c0 + 1];
    float w10 = We[1 * HIDDEN + c0], w11 = We[1 * HIDDEN + c0 + 1];
    float w20 = We[2 * HIDDEN + c0], w21 = We[2 * HIDDEN + c0 + 1];
    float w30 = We[3 * HIDDEN + c0], w31 = We[3 * HIDDEN + c0 + 1];
    float bb0 = be[c0], bb1 = be[c0 + 1];

    int e = wave;
    bool valid = (e < NEDGES);
    int src = 0, dst = 0;
    float4 a; a.x = a.y = a.z = a.w = 0.f;
    if (valid) {
        src = (int)ei[e];
        dst = (int)ei[NEDGES + e];
        a   = ((const float4*)ea)[e];
    }

    while (valid) {
        const int  en = e + nwv;
        const bool vn = (en < NEDGES);
        int srcn = 0, dstn = 0;
        float4 an; an.x = an.y = an.z = an.w = 0.f;
        if (vn) {
            srcn = (int)ei[en];
            dstn = (int)ei[NEDGES + en];
            an   = ((const float4*)ea)[en];
            // pull next gather row toward the WGP while we work
            __builtin_prefetch(x + (size_t)srcn * HIDDEN + c0, 0, 3);
        }

        float e0 = bb0 + a.x * w00 + a.y * w10 + a.z * w20 + a.w * w30;
        float e1 = bb1 + a.x * w01 + a.y * w11 + a.z * w21 + a.w * w31;

        float2 xs = ((const float2*)(x + (size_t)src * HIDDEN))[lane];
        float m0 = xs.x + e0; m0 = m0 > 0.f ? m0 : 0.f;
        float m1 = xs.y + e1; m1 = m1 > 0.f ? m1 : 0.f;

        float* p = agg + (size_t)dst * HIDDEN + c0;
        atomicAdd(p,     m0);                       // global_atomic_add_f32
        atomicAdd(p + 1, m1);

        e = en; src = srcn; dst = dstn; a = an; valid = vn;
    }
}

// ---------------- kernel 2: H = x + agg, then WMMA MLP -----------------
// 128 threads = 4 waves; each wave owns a 16-node tile.
// Weights staged in LDS in *fragment-pair* layout so each WMMA B-operand
// is one aligned ds_load_b64 straight into an even VGPR pair:
//   sWp[ ((k>>1)*64 + n)*2 + (k&1) ] = W[k][n]
// B fragment for K-step ks=4*kk+2*half is pair id m = 2*kk+half.
__global__ void __launch_bounds__(128)
gine_mlp_kernel(const float* __restrict__ x,
                const float* __restrict__ agg,
                const float* __restrict__ W1, const float* __restrict__ b1,
                const float* __restrict__ W2, const float* __restrict__ b2,
                float* __restrict__ out)
{
    __shared__ float sW1p[HIDDEN * HIDDEN];  // 16 KB, pair layout
    __shared__ float sW2p[HIDDEN * HIDDEN];  // 16 KB, pair layout
    __shared__ float sb1[HIDDEN];
    __shared__ float sb2[HIDDEN];
    __shared__ float sT[4][16 * HIDDEN];     // per-wave relu intermediate, 16 KB

    const int tid = threadIdx.x;

    // stage weights directly into pair layout (coalesced b32 reads from L2)
    for (int i = tid; i < 2048; i += 128) {
        int m = i >> 6, n = i & 63;
        v2f p1, p2;
        p1.x = W1[(2 * m) * HIDDEN + n];
        p1.y = W1[(2 * m + 1) * HIDDEN + n];
        p2.x = W2[(2 * m) * HIDDEN + n];
        p2.y = W2[(2 * m + 1) * HIDDEN + n];
        ((v2f*)sW1p)[i] = p1;
        ((v2f*)sW2p)[i] = p2;
    }
    if (tid < HIDDEN) { sb1[tid] = b1[tid]; sb2[tid] = b2[tid]; }
    __syncthreads();

    const int wave = tid >> 5;
    const int lane = tid & 31;
    const int half = lane >> 4;      // 0: lanes 0-15, 1: lanes 16-31
    const int l    = lane & 15;
    const int row0 = (blockIdx.x * 4 + wave) * 16;

    const v2f* W1f = (const v2f*)sW1p;
    const v2f* W2f = (const v2f*)sW2p;

    // ---- A fragments of H = x + agg (clamped row for tail tiles) ----
    int gr = row0 + l; if (gr > NNODES - 1) gr = NNODES - 1;
    const float* xr = x   + (size_t)gr * HIDDEN;
    const float* ar = agg + (size_t)gr * HIDDEN;

    v2f afrag[16];
#pragma unroll
    for (int kk = 0; kk < 16; ++kk) {
        int col = 4 * kk + 2 * half;
        float2 hx = *(const float2*)(xr + col);
        float2 ha = *(const float2*)(ar + col);
        afrag[kk].x = hx.x + ha.x;
        afrag[kk].y = hx.y + ha.y;
    }

    float* tw = sT[wave];

    // ---- GEMM1: t = relu(H @ W1 + b1) ----
#pragma unroll
    for (int nt = 0; nt < 4; ++nt) {
        v8f c = {};
#pragma unroll
        for (int kk = 0; kk < 16; ++kk) {
            v2f bfr = W1f[(2 * kk + half) * HIDDEN + nt * 16 + l];
            c = __builtin_amdgcn_wmma_f32_16x16x4_f32(
                    false, afrag[kk], false, bfr, (short)0, c, false, false);
        }
        float bias = sb1[nt * 16 + l];
#pragma unroll
        for (int j = 0; j < 8; ++j) {
            float v = c[j] + bias;
            v = v > 0.f ? v : 0.f;
            tw[(j + 8 * half) * HIDDEN + nt * 16 + l] = v;  // D-layout -> row-major
        }
    }

    // ---- reload A fragments of t from LDS (same-wave DS ops in order) ----
#pragma unroll
    for (int kk = 0; kk < 16; ++kk) {
        int col = 4 * kk + 2 * half;
        float2 t2 = *(const float2*)(tw + l * HIDDEN + col);
        afrag[kk].x = t2.x;
        afrag[kk].y = t2.y;
    }

    // ---- GEMM2: out = t @ W2 + b2 ----
#pragma unroll
    for (int nt = 0; nt < 4; ++nt) {
        v8f c = {};
#pragma unroll
        for (int kk = 0; kk < 16; ++kk) {
            v2f bfr = W2f[(2 * kk + half) * HIDDEN + nt * 16 + l];
            c = __builtin_amdgcn_wmma_f32_16x16x4_f32(
                    false, afrag[kk], false, bfr, (short)0, c, false, false);
        }
        float bias = sb2[nt * 16 + l];
#pragma unroll
        for (int j = 0; j < 8; ++j) {
            int grow = row0 + j + 8 * half;
            if (grow < NNODES)
                out[(size_t)grow * HIDDEN + nt * 16 + l] = c[j] + bias;
        }
    }
}

// ------------------------------- launch --------------------------------
extern "C" void kernel_launch(void* const* d_in, const int* in_sizes, int n_in,
                              void* d_out, int out_size, void* d_ws, size_t ws_size,
                              hipStream_t stream)
{
    const float*     x  = (const float*)d_in[0];
    const long long* ei = (const long long*)d_in[1];   // int64 [2,E]
    const float*     ea = (const float*)d_in[2];
    const float*     We = (const float*)d_in[3];
    const float*     be = (const float*)d_in[4];
    const float*     W1 = (const float*)d_in[5];
    const float*     b1 = (const float*)d_in[6];
    const float*     W2 = (const float*)d_in[7];
    const float*     b2 = (const float*)d_in[8];
    float* out = (float*)d_out;
    float* agg = (float*)d_ws;                         // [N, 64] f32 = 25.6 MB

    const int n4 = NNODES * HIDDEN / 4;
    gine_zero_kernel<<<(n4 + 255) / 256, 256, 0, stream>>>(agg, n4);

    // 8192 blocks * 8 waves = 65536 waves, ~24 edges/wave grid-stride
    gine_edge_kernel<<<8192, 256, 0, stream>>>(x, ei, ea, We, be, agg);

    gine_mlp_kernel<<<(NNODES + 63) / 64, 128, 0, stream>>>(
        x, agg, W1, b1, W2, b2, out);
}